// LatentMemory_54958401519767
// MI455X (gfx1250) — compile-verified
//
#include <hip/hip_runtime.h>
#include <hip/hip_bf16.h>

#define BATCH    8192
#define FEAT     512
#define NDATA    500000
#define NCLASS   10000
#define MOM      0.9f

#if __has_builtin(__builtin_amdgcn_global_load_async_to_lds_b128) && \
    __has_builtin(__builtin_amdgcn_s_wait_asynccnt)
#define USE_ASYNC_LDS 1
#else
#define USE_ASYNC_LDS 0
#endif

#if __has_builtin(__builtin_amdgcn_sched_barrier)
#define SCHED_FENCE() __builtin_amdgcn_sched_barrier(0)
#else
#define SCHED_FENCE()
#endif

#if USE_ASYNC_LDS
// Builtin parameter types (probe-confirmed by round-1 diagnostic):
//   arg0: int __vector_size__(16) addrspace(1)*  (global src)
//   arg1: int __vector_size__(16) addrspace(3)*  (LDS dst)
typedef int v4i_t __attribute__((__vector_size__(16)));
typedef __attribute__((address_space(1))) v4i_t gv4i_t;
typedef __attribute__((address_space(3))) v4i_t lv4i_t;
#endif

// One workgroup per batch row: gather lat_memory[idx[row]] (async -> LDS),
// blend with batch_samples[row], L2-normalize via wave32 shuffle reduction,
// scatter into out_lat, and fp32-atomic segment-sum into out_cs.
__global__ __launch_bounds__(128) void lat_update_kernel(
    const float* __restrict__ batch,    // [BATCH, FEAT]
    const float* __restrict__ latmem,   // [NDATA, FEAT]
    const int*   __restrict__ targets,  // [BATCH]
    const int*   __restrict__ idx,      // [BATCH]
    float*       __restrict__ out_lat,  // [NDATA, FEAT]
    float*       __restrict__ out_cs)   // [NCLASS, FEAT]
{
    const int row  = blockIdx.x;
    const int tid  = threadIdx.x;        // 0..127, 4 floats each
    const int lane = tid & 31;
    const int wave = tid >> 5;

    const int j = idx[row];              // unique per row (permutation)
    const int t = targets[row];

    const float* gsrc = latmem + (size_t)j   * FEAT + tid * 4;
    const float* bsrc = batch  + (size_t)row * FEAT + tid * 4;
    float*       csum = out_cs + (size_t)t   * FEAT + tid * 4;

    // warm L2 for the atomic-add target row (gfx1250 global_prefetch path)
    __builtin_prefetch(csum, 1, 3);

    __shared__ float lds_row[FEAT];
    __shared__ float red[4];

    float4 lv, bv;
#if USE_ASYNC_LDS
    // Issue the streaming batch load FIRST (LOADcnt), fence the scheduler so
    // it cannot sink that load below s_wait_asynccnt, then issue the
    // dependent async gather (ASYNCcnt). Both loads are then in flight while
    // the wave waits only on the async one.
    bv = *(const float4*)bsrc;                       // global_load_b128
    SCHED_FENCE();
    __builtin_amdgcn_global_load_async_to_lds_b128(
        (gv4i_t*)gsrc, (lv4i_t*)&lds_row[tid * 4], 0, 0);
    __builtin_amdgcn_s_wait_asynccnt(0);             // s_wait_asynccnt 0
    lv = *(const float4*)&lds_row[tid * 4];          // ds_load_b128
#else
    lv = *(const float4*)gsrc;
    bv = *(const float4*)bsrc;
#endif

    // blended = lat*(1-M) + batch*M
    float4 u;
    u.x = fmaf(lv.x, 1.0f - MOM, bv.x * MOM);
    u.y = fmaf(lv.y, 1.0f - MOM, bv.y * MOM);
    u.z = fmaf(lv.z, 1.0f - MOM, bv.z * MOM);
    u.w = fmaf(lv.w, 1.0f - MOM, bv.w * MOM);

    // block-wide sum of squares: wave32 shuffle tree + cross-wave LDS
    float p = u.x * u.x + u.y * u.y + u.z * u.z + u.w * u.w;
#pragma unroll
    for (int off = 16; off > 0; off >>= 1)
        p += __shfl_xor(p, off, 32);
    if (lane == 0) red[wave] = p;
    __syncthreads();
    const float s   = red[0] + red[1] + red[2] + red[3];
    const float inv = 1.0f / sqrtf(s);

    u.x *= inv; u.y *= inv; u.z *= inv; u.w *= inv;

    // scatter normalized row into the copied memory bank (unique j -> no race)
    *(float4*)(out_lat + (size_t)j * FEAT + tid * 4) = u;

    // per-class segment sum: hardware fp32 global atomics (low contention,
    // 20.5MB region is L2-resident on MI455X's 192MB L2)
    unsafeAtomicAdd(csum + 0, u.x);
    unsafeAtomicAdd(csum + 1, u.y);
    unsafeAtomicAdd(csum + 2, u.z);
    unsafeAtomicAdd(csum + 3, u.w);
}

extern "C" void kernel_launch(void* const* d_in, const int* in_sizes, int n_in,
                              void* d_out, int out_size, void* d_ws, size_t ws_size,
                              hipStream_t stream) {
    const float* batch   = (const float*)d_in[0];  // batch_samples
    const float* latmem  = (const float*)d_in[1];  // lat_memory
    const float* csums   = (const float*)d_in[2];  // class_sums
    const int*   targets = (const int*)d_in[3];
    const int*   idx     = (const int*)d_in[4];

    float* out_lat = (float*)d_out;                       // [NDATA, FEAT]
    float* out_cs  = out_lat + (size_t)NDATA * FEAT;      // [NCLASS, FEAT]

    // Bulk D2D copies saturate HBM (the 2.05GB here is the roofline floor);
    // stream order guarantees the update kernel overwrites after the copy.
    (void)hipMemcpyAsync(out_lat, latmem, (size_t)NDATA * FEAT * sizeof(float),
                         hipMemcpyDeviceToDevice, stream);
    (void)hipMemcpyAsync(out_cs, csums, (size_t)NCLASS * FEAT * sizeof(float),
                         hipMemcpyDeviceToDevice, stream);

    lat_update_kernel<<<BATCH, 128, 0, stream>>>(batch, latmem, targets, idx,
                                                 out_lat, out_cs);
}